// JKNet_GATConv_35802847379840
// MI455X (gfx1250) — compile-verified
//
#include <hip/hip_runtime.h>
#include <hip/hip_bf16.h>

// ---------------------------------------------------------------------------
// JKNet-GAT (3 layers, 4 heads x 64, JK-cat, linear head) for MI455X (gfx1250)
// - Dense GEMMs on the WMMA pipe: V_WMMA_F32_16X16X4_F32 (fp32-exact vs ref).
// - GEMM stages tiles with GLOBAL_LOAD_ASYNC_TO_LDS_B128 (ASYNCcnt), double
//   buffered so the async DMA of chunk k+1 overlaps WMMA math on chunk k.
// - Edge softmax/aggregation uses f32 global atomics (the real bottleneck:
//   ~1.7 GB irregular traffic per layer vs ~4 us for the dense GEMM reads).
// ---------------------------------------------------------------------------

typedef __attribute__((ext_vector_type(2))) float v2f;
typedef __attribute__((ext_vector_type(8))) float v8f;

#define N_FEAT  500
#define HIDDEN  256
#define N_HEADS 4
#define N_CLASS 40

// ---- CDNA5 async Global->LDS support (guarded so the build stays green) ----
#if defined(__has_builtin)
#  if __has_builtin(__builtin_amdgcn_global_load_async_to_lds_b128)
#    define HAVE_ASYNC_LDS 1
#  endif
#  if __has_builtin(__builtin_amdgcn_s_wait_asynccnt)
#    define HAVE_WAIT_ASYNC 1
#  endif
#endif
#ifndef HAVE_ASYNC_LDS
#  define HAVE_ASYNC_LDS 0
#endif
#ifndef HAVE_WAIT_ASYNC
#  define HAVE_WAIT_ASYNC 0
#endif

#if HAVE_ASYNC_LDS
// Builtin signature (from clang diagnostics):
//   void (v4i AS1* gsrc, v4i AS3* ldst, imm offset, imm cpol)
// where v4i = int __attribute__((vector_size(16))).  generic->AS1 is
// numerically identity; generic LDS addr[31:0] is the LDS byte offset
// (flat aperture rule), so inttoptr truncation yields the AS3 address.
typedef int v4i_vs __attribute__((vector_size(16)));
typedef __attribute__((address_space(1))) v4i_vs* as1_v4i;
typedef __attribute__((address_space(3))) v4i_vs* as3_v4i;

__device__ __forceinline__ as1_v4i gbl_cast(const void* p) {
  return (as1_v4i)(uintptr_t)p;
}
__device__ __forceinline__ as3_v4i lds_cast(void* p) {
  return (as3_v4i)(unsigned int)(uintptr_t)p;
}
#endif

__device__ __forceinline__ void wait_async0() {
#if HAVE_WAIT_ASYNC
  __builtin_amdgcn_s_wait_asynccnt(0);
#elif HAVE_ASYNC_LDS
  asm volatile("s_wait_asynccnt 0x0" ::: "memory");
#endif
}

__device__ __forceinline__ v8f wmma4(v2f a, v2f b, v8f c) {
  // D = A(16x4,f32) * B(4x16,f32) + C(16x16,f32), wave32
  return __builtin_amdgcn_wmma_f32_16x16x4_f32(
      /*neg_a=*/false, a, /*neg_b=*/false, b,
      /*c_mod=*/(short)0, c, /*reuse_a=*/false, /*reuse_b=*/false);
}

// ---------------- Tiled WMMA GEMM: Z[M,N] = X[M,K](row stride ldx) @ W[K,N]
// block = 256 thr (8 waves). Tile: BM=128 rows (16/wave), BN=64 cols, KC=16.
#define BM 128
#define BN 64
#define KC 16
#define AS_STRIDE 20   // 80B rows: B128-aligned; 20r mod 64 distinct (r<16)
#define BS_STRIDE 80   // 16k+n banking: half-waves hit disjoint bank sets
#define A_TILE (BM * AS_STRIDE)
#define B_TILE (KC * BS_STRIDE)

__global__ __launch_bounds__(256) void gat_gemm_wmma(
    const float* __restrict__ X, int ldx,
    const float* __restrict__ W, float* __restrict__ Z,
    int M, int N, int K) {
  __shared__ float As[2 * A_TILE];
  __shared__ float Bs[2 * B_TILE];

  const int tid  = threadIdx.x;
  const int wave = tid >> 5;
  const int lane = tid & 31;
  const int rIW  = lane & 15;          // row/col-in-tile index
  const int koff = (lane >> 4) << 1;   // 0 or 2 (upper half-wave K offset)
  const int bm   = blockIdx.x * BM;
  const int bn   = blockIdx.y * BN;

  v8f acc[4];
#pragma unroll
  for (int t = 0; t < 4; ++t)
#pragma unroll
    for (int r = 0; r < 8; ++r) acc[t][r] = 0.0f;

  // Stage one (A,B) K-chunk into buffer `buf`. Rows >= kc / >= M are simply
  // not written: the ks-loop never reads rows >= kc, and garbage A rows only
  // feed D rows whose stores are masked.
  auto stage = [&](int kk, int buf) {
    float* as = &As[buf * A_TILE];
    float* bs = &Bs[buf * B_TILE];
    const int kc = (K - kk < KC) ? (K - kk) : KC;
    for (int q = tid; q < BM * 4; q += 256) {
      const int r  = q >> 2;
      const int k4 = (q & 3) << 2;
      const int gr = bm + r;
      if (gr < M && k4 < kc) {
#if HAVE_ASYNC_LDS
        __builtin_amdgcn_global_load_async_to_lds_b128(
            gbl_cast(X + (size_t)gr * ldx + kk + k4),
            lds_cast(&as[r * AS_STRIDE + k4]), 0, 0);
#else
        const float4 v = *(const float4*)(X + (size_t)gr * ldx + kk + k4);
        float* a = &as[r * AS_STRIDE + k4];
        a[0] = v.x; a[1] = v.y; a[2] = v.z; a[3] = v.w;
#endif
      }
    }
    const int r  = tid >> 4;
    const int n4 = (tid & 15) << 2;
    if (r < kc) {
#if HAVE_ASYNC_LDS
      __builtin_amdgcn_global_load_async_to_lds_b128(
          gbl_cast(W + (size_t)(kk + r) * N + bn + n4),
          lds_cast(&bs[r * BS_STRIDE + n4]), 0, 0);
#else
      const float4 v = *(const float4*)(W + (size_t)(kk + r) * N + bn + n4);
      float* b = &bs[r * BS_STRIDE + n4];
      b[0] = v.x; b[1] = v.y; b[2] = v.z; b[3] = v.w;
#endif
    }
  };

  stage(0, 0);
  wait_async0();
  __syncthreads();

  int cur = 0;
  for (int kk = 0; kk < K; kk += KC) {
    const int kc = (K - kk < KC) ? (K - kk) : KC;   // K%4==0 always here
    if (kk + KC < K) stage(kk + KC, cur ^ 1);       // DMA next chunk (overlap)

    const float* as = &As[cur * A_TILE];
    const float* bs = &Bs[cur * B_TILE];
    for (int ks = 0; ks < kc; ks += 4) {
      v2f a;
      const float* ap = &as[(wave * 16 + rIW) * AS_STRIDE + ks + koff];
      a.x = ap[0]; a.y = ap[1];
#pragma unroll
      for (int t = 0; t < 4; ++t) {
        v2f b;
        b.x = bs[(ks + koff)     * BS_STRIDE + t * 16 + rIW];
        b.y = bs[(ks + koff + 1) * BS_STRIDE + t * 16 + rIW];
        acc[t] = wmma4(a, b, acc[t]);
      }
    }
    wait_async0();
    __syncthreads();
    cur ^= 1;
  }

  // D layout: acc[t][r] = D[r + 8*(lane/16)][lane%16] of n-tile t.
  const int mtile = bm + wave * 16;
  const int mbase = mtile + ((lane >> 4) << 3);
  const int nbase = bn + rIW;
  if (mtile + 16 <= M) {            // wave-uniform fast path: no exec juggling
#pragma unroll
    for (int t = 0; t < 4; ++t)
#pragma unroll
      for (int r = 0; r < 8; ++r)
        Z[(size_t)(mbase + r) * N + nbase + t * 16] = acc[t][r];
  } else if (mtile < M) {           // partial tile (unused for M=50000)
#pragma unroll
    for (int t = 0; t < 4; ++t)
#pragma unroll
      for (int r = 0; r < 8; ++r) {
        const int gm = mbase + r;
        if (gm < M) Z[(size_t)gm * N + nbase + t * 16] = acc[t][r];
      }
  }
}

// ---------------- Head GEMM: Out[M,40] = H[M,768] @ Wo[768,40] + bo
// One wave per 16-row tile; N padded to 48 (3 n-tiles) with masked B/stores.
__global__ __launch_bounds__(128) void gat_out_gemm_wmma(
    const float* __restrict__ H, const float* __restrict__ Wo,
    const float* __restrict__ bo, float* __restrict__ Out,
    int M, int K, int N) {
  const int lane = threadIdx.x & 31;
  const int wave = threadIdx.x >> 5;
  const int tile = blockIdx.x * 4 + wave;
  if (tile * 16 >= M) return;                 // wave-uniform exit
  const int m0   = tile * 16;
  const int rIW  = lane & 15;
  const int koff = (lane >> 4) << 1;

  v8f acc[3];
#pragma unroll
  for (int t = 0; t < 3; ++t)
#pragma unroll
    for (int r = 0; r < 8; ++r) acc[t][r] = 0.0f;

  const float* arow = H + (size_t)(m0 + rIW) * K;
  for (int k = 0; k < K; k += 4) {
    const float2 av = *(const float2*)(arow + k + koff);
    v2f a; a.x = av.x; a.y = av.y;
#pragma unroll
    for (int t = 0; t < 3; ++t) {
      const int n = t * 16 + rIW;
      v2f b;
      b.x = (n < N) ? Wo[(size_t)(k + koff)     * N + n] : 0.f;
      b.y = (n < N) ? Wo[(size_t)(k + koff + 1) * N + n] : 0.f;
      acc[t] = wmma4(a, b, acc[t]);
    }
  }
  const int mb = m0 + ((lane >> 4) << 3);
#pragma unroll
  for (int t = 0; t < 3; ++t) {
    const int n = t * 16 + rIW;
    if (n < N) {
      const float bias = bo[n];
#pragma unroll
      for (int r = 0; r < 8; ++r)
        Out[(size_t)(mb + r) * N + n] = acc[t][r] + bias;
    }
  }
}

// ---------------- attention helper kernels ----------------

__global__ void init_buffers(float* __restrict__ agg, float* __restrict__ m,
                             float* __restrict__ denom, int n) {
  const int i = blockIdx.x * blockDim.x + threadIdx.x;
  if (i < n * HIDDEN) agg[i] = 0.f;
  if (i < n * N_HEADS) { m[i] = -1e30f; denom[i] = 0.f; }
}

// es[n][h] = sum_c z[n][h*64+c]*a_src[h*64+c]; one wave per node,
// 8 contiguous features/lane (head = lane/8), 3-step shfl_xor reduce.
__global__ void attn_scores(const float* __restrict__ Z,
                            const float* __restrict__ a_src,
                            const float* __restrict__ a_dst,
                            float* __restrict__ es, float* __restrict__ ed,
                            int n) {
  const int w    = blockIdx.x * (blockDim.x >> 5) + (threadIdx.x >> 5);
  const int lane = threadIdx.x & 31;
  if (w >= n) return;                        // wave-uniform
  const float* z = Z + (size_t)w * HIDDEN;
  const int base = lane * 8;
  float ss = 0.f, sd = 0.f;
#pragma unroll
  for (int j = 0; j < 8; ++j) {
    const float zz = z[base + j];
    ss += zz * a_src[base + j];
    sd += zz * a_dst[base + j];
  }
  ss += __shfl_xor(ss, 1, 32); sd += __shfl_xor(sd, 1, 32);
  ss += __shfl_xor(ss, 2, 32); sd += __shfl_xor(sd, 2, 32);
  ss += __shfl_xor(ss, 4, 32); sd += __shfl_xor(sd, 4, 32);
  if ((lane & 7) == 0) {
    const int h = lane >> 3;
    es[(size_t)w * N_HEADS + h] = ss;
    ed[(size_t)w * N_HEADS + h] = sd;
  }
}

__device__ __forceinline__ void atomicMaxFloat(float* addr, float val) {
  unsigned int* u = (unsigned int*)addr;
  unsigned int old = *u;
  while (true) {
    if (__uint_as_float(old) >= val) break;
    const unsigned int assumed = old;
    old = atomicCAS(u, assumed, __float_as_uint(val));
    if (old == assumed) break;
  }
}

// pass 1: e = leaky_relu(es[src]+ed[dst], 0.2); segment max into m[dst]
__global__ void edge_logits(const int* __restrict__ src, const int* __restrict__ dst,
                            int E, int n,
                            const float* __restrict__ es, const float* __restrict__ ed,
                            float* __restrict__ e, float* __restrict__ m) {
  const int i = blockIdx.x * blockDim.x + threadIdx.x;
  const int total = E + n;                   // self-loops appended
  if (i >= total) return;
  const int s = (i < E) ? src[i] : (i - E);
  const int d = (i < E) ? dst[i] : (i - E);
#pragma unroll
  for (int h = 0; h < N_HEADS; ++h) {
    float v = es[(size_t)s * N_HEADS + h] + ed[(size_t)d * N_HEADS + h];
    v = (v > 0.f) ? v : 0.2f * v;
    e[(size_t)i * N_HEADS + h] = v;
    atomicMaxFloat(&m[(size_t)d * N_HEADS + h], v);
  }
}

// pass 2: ex = exp(e - m[dst]); denom[dst] += ex (e overwritten with ex)
__global__ void edge_exp(const int* __restrict__ dst, int E, int n,
                         float* __restrict__ e, const float* __restrict__ m,
                         float* __restrict__ denom) {
  const int i = blockIdx.x * blockDim.x + threadIdx.x;
  const int total = E + n;
  if (i >= total) return;
  const int d = (i < E) ? dst[i] : (i - E);
#pragma unroll
  for (int h = 0; h < N_HEADS; ++h) {
    const float ex = __expf(e[(size_t)i * N_HEADS + h] - m[(size_t)d * N_HEADS + h]);
    e[(size_t)i * N_HEADS + h] = ex;
    atomicAdd(&denom[(size_t)d * N_HEADS + h], ex);
  }
}

// pass 3: agg[dst] += (ex/denom[dst]) * z[src]; one wave per edge,
// coalesced 256-feature gather + scatter (8 features/lane).
__global__ void edge_aggregate(const int* __restrict__ src, const int* __restrict__ dst,
                               int E, int n,
                               const float* __restrict__ e, const float* __restrict__ denom,
                               const float* __restrict__ Z, float* __restrict__ agg) {
  const int wv   = blockIdx.x * (blockDim.x >> 5) + (threadIdx.x >> 5);
  const int lane = threadIdx.x & 31;
  const int total = E + n;
  if (wv >= total) return;                   // wave-uniform
  const int s = (wv < E) ? src[wv] : (wv - E);
  const int d = (wv < E) ? dst[wv] : (wv - E);
  const float* zs = Z   + (size_t)s * HIDDEN;
  float*       od = agg + (size_t)d * HIDDEN;
#pragma unroll
  for (int j = 0; j < 8; ++j) {
    const int c = lane + j * 32;             // 0..255 coalesced
    const int h = c >> 6;
    const float alpha = e[(size_t)wv * N_HEADS + h] /
                        (denom[(size_t)d * N_HEADS + h] + 1e-16f);
    atomicAdd(&od[c], alpha * zs[c]);
  }
}

// epilogue: h = relu(agg + b) written into the JK-cat slice (row stride 768)
__global__ void bias_relu_store(const float* __restrict__ agg, const float* __restrict__ b,
                                float* __restrict__ h, int ldh, int n) {
  const long long i = (long long)blockIdx.x * blockDim.x + threadIdx.x;
  if (i >= (long long)n * HIDDEN) return;
  const int node = (int)(i >> 8);
  const int c    = (int)(i & (HIDDEN - 1));
  const float v = agg[i] + b[c];
  h[(size_t)node * ldh + c] = (v > 0.f) ? v : 0.f;
}

// ---------------------------------------------------------------------------
extern "C" void kernel_launch(void* const* d_in, const int* in_sizes, int n_in,
                              void* d_out, int out_size, void* d_ws, size_t ws_size,
                              hipStream_t stream) {
  (void)n_in; (void)out_size; (void)ws_size;
  const float* x    = (const float*)d_in[0];
  const int*   ei   = (const int*)d_in[1];
  const float* W[3]   = {(const float*)d_in[2], (const float*)d_in[6],  (const float*)d_in[10]};
  const float* as_[3] = {(const float*)d_in[3], (const float*)d_in[7],  (const float*)d_in[11]};
  const float* ad_[3] = {(const float*)d_in[4], (const float*)d_in[8],  (const float*)d_in[12]};
  const float* bb[3]  = {(const float*)d_in[5], (const float*)d_in[9],  (const float*)d_in[13]};
  const float* Wout = (const float*)d_in[14];
  const float* bout = (const float*)d_in[15];
  float* out = (float*)d_out;

  const int n = in_sizes[0] / N_FEAT;   // 50000
  const int E = in_sizes[1] / 2;        // 800000
  const int* src = ei;
  const int* dst = ei + E;
  const int total = E + n;              // + self-loops

  // ---- workspace carve ----
  char* ws = (char*)d_ws;
  size_t off = 0;
  auto carve = [&](size_t nf) -> float* {
    float* p = (float*)(ws + off);
    off += ((nf * sizeof(float)) + 255) & ~(size_t)255;
    return p;
  };
  float* hcat = carve((size_t)n * HIDDEN * 3);   // 153.6 MB JK-cat buffer
  float* Zb   = carve((size_t)n * HIDDEN);       // z = x@W
  float* agg  = carve((size_t)n * HIDDEN);       // attention-weighted sum
  float* ebuf = carve((size_t)total * N_HEADS);  // edge logits / exp
  float* es   = carve((size_t)n * N_HEADS);
  float* ed   = carve((size_t)n * N_HEADS);
  float* mbuf = carve((size_t)n * N_HEADS);
  float* dbuf = carve((size_t)n * N_HEADS);

  for (int l = 0; l < 3; ++l) {
    const float* Xin = (l == 0) ? x : (hcat + (size_t)(l - 1) * HIDDEN);
    const int K   = (l == 0) ? N_FEAT : HIDDEN;
    const int ldx = (l == 0) ? N_FEAT : HIDDEN * 3;

    dim3 ggrid((n + BM - 1) / BM, HIDDEN / BN);
    gat_gemm_wmma<<<ggrid, 256, 0, stream>>>(Xin, ldx, W[l], Zb, n, HIDDEN, K);

    init_buffers<<<((size_t)n * HIDDEN + 255) / 256, 256, 0, stream>>>(agg, mbuf, dbuf, n);
    attn_scores<<<(n + 7) / 8, 256, 0, stream>>>(Zb, as_[l], ad_[l], es, ed, n);
    edge_logits<<<(total + 255) / 256, 256, 0, stream>>>(src, dst, E, n, es, ed, ebuf, mbuf);
    edge_exp<<<(total + 255) / 256, 256, 0, stream>>>(dst, E, n, ebuf, mbuf, dbuf);
    edge_aggregate<<<(total + 7) / 8, 256, 0, stream>>>(src, dst, E, n, ebuf, dbuf, Zb, agg);
    bias_relu_store<<<((size_t)n * HIDDEN + 255) / 256, 256, 0, stream>>>(
        agg, bb[l], hcat + (size_t)l * HIDDEN, HIDDEN * 3, n);
  }

  const int tiles = n / 16;                      // 3125 (exact)
  gat_out_gemm_wmma<<<(tiles + 3) / 4, 128, 0, stream>>>(
      hcat, Wout, bout, out, n, HIDDEN * 3, N_CLASS);
}